// MessagePassingLayer_28346784154211
// MI455X (gfx1250) — compile-verified
//
#include <hip/hip_runtime.h>

// CompGCN-style message passing layer for MI455X (gfx1250).
// Linearity trick: matmul commutes with segment_sum -> scatter composed
// features (hw f32 atomics; x fits in 192MB L2), then 3 dense fp32 WMMA
// GEMMs [100000x256]x[256x256]. Each wave computes a 16x64 strip (4 n-tiles)
// so the A operand is read once per 4 output tiles (4x less A traffic).
// Workspace: acc[N*D] + deg[N] + sums[1024] floats ~= 102.8 MB.

#define N_NODES 100000
#define N_REL   474
#define DIM     256
#define N_EDGES 1000000
#define BN_EPS  1e-5f
#define NT      4        // 16-wide n-tiles per wave (64 columns)

typedef __attribute__((ext_vector_type(2))) float v2f;
typedef __attribute__((ext_vector_type(8))) float v8f;

__global__ void zero_f32(float* __restrict__ p, long n) {
  long i = (long)blockIdx.x * blockDim.x + threadIdx.x;
  long stride = (long)gridDim.x * blockDim.x;
  for (; i < n; i += stride) p[i] = 0.0f;
}

__global__ void degree_kernel(const int* __restrict__ row, float* __restrict__ deg, int e) {
  int i = blockIdx.x * blockDim.x + threadIdx.x;
  if (i < e) unsafeAtomicAdd(&deg[row[i]], 1.0f);
}

__global__ void deginv_kernel(float* __restrict__ deg, int n) {
  int i = blockIdx.x * blockDim.x + threadIdx.x;
  if (i < n) { float d = deg[i]; deg[i] = d > 0.f ? rsqrtf(d) : 0.f; }
}

// 64 threads per edge, float4 per thread: acc[row] += norm*(x[col] - r[et])
__global__ void scatter_kernel(const float* __restrict__ x, const float* __restrict__ r,
                               const int* __restrict__ ei, const int* __restrict__ et,
                               const float* __restrict__ dinv, float* __restrict__ acc,
                               int e0, int ecount) {
  long t = (long)blockIdx.x * blockDim.x + threadIdx.x;
  int eidx = (int)(t >> 6);
  if (eidx >= ecount) return;
  int e  = e0 + eidx;
  int f  = ((int)t & 63) * 4;
  int rw = ei[e];
  int cl = ei[N_EDGES + e];
  int ty = et[e];
  float nrm = dinv[rw] * dinv[cl];
  const float4 xv = *(const float4*)(x + (long)cl * DIM + f);
  const float4 rv = *(const float4*)(r + (long)ty * DIM + f);
  float* dst = acc + (long)rw * DIM + f;
  unsafeAtomicAdd(dst + 0, nrm * (xv.x - rv.x));
  unsafeAtomicAdd(dst + 1, nrm * (xv.y - rv.y));
  unsafeAtomicAdd(dst + 2, nrm * (xv.z - rv.z));
  unsafeAtomicAdd(dst + 3, nrm * (xv.w - rv.w));
}

// Each wave computes a 16x64 strip (NT=4 n-tiles), K=256 via
// 64 K-steps x NT V_WMMA_F32_16X16X4_F32. A loaded once per K-step.
// MODE 0: C = (A - loop_rel_bcast) @ B + bias   (write)
// MODE 1: C += A @ B                            (accumulate into existing C)
// MODE 2: C = A @ B with M-row guard            (write, 474-row rel GEMM)
template <int MODE>
__global__ void gemm_kernel(const float* __restrict__ A, const float* __restrict__ B,
                            const float* __restrict__ loop_rel, const float* __restrict__ bias,
                            float* __restrict__ C, int M) {
  const int NGRP = DIM / (16 * NT);        // 4 n-groups of 64 columns
  int wave = (int)((blockIdx.x * blockDim.x + threadIdx.x) >> 5);
  int lane = threadIdx.x & 31;
  int mt = wave / NGRP;
  int nbase = (wave % NGRP) * (16 * NT);   // first column of this wave's strip
  int m0 = mt * 16;
  int lm = lane & 15;                      // row (A) / col (B,C) within tile
  int hi = lane >> 4;                      // 0 or 1
  int lk = hi * 2;                         // K sub-offset for A/B operands

  v8f c[NT];
  if (MODE == 1) {
#pragma unroll
    for (int j = 0; j < NT; ++j)
#pragma unroll
      for (int i = 0; i < 8; ++i) {
        int row = m0 + i + hi * 8;
        c[j][i] = C[(long)row * DIM + nbase + j * 16 + lm];
      }
  } else {
#pragma unroll
    for (int j = 0; j < NT; ++j)
#pragma unroll
      for (int i = 0; i < 8; ++i) c[j][i] = 0.0f;
  }

  int arow = m0 + lm;
  bool avalid = (MODE != 2) || (arow < M);
  const float* Arow = A + (long)arow * DIM;
  const float* Bcol = B + nbase + lm;      // column base for this lane

#pragma unroll 2
  for (int k0 = 0; k0 < DIM; k0 += 4) {
    int ka = k0 + lk;
    v2f a;
    if (MODE == 0) {
      a[0] = Arow[ka]     - loop_rel[ka];
      a[1] = Arow[ka + 1] - loop_rel[ka + 1];
    } else if (MODE == 2) {
      a[0] = avalid ? Arow[ka]     : 0.0f;
      a[1] = avalid ? Arow[ka + 1] : 0.0f;
    } else {
      a[0] = Arow[ka];
      a[1] = Arow[ka + 1];
    }
#pragma unroll
    for (int j = 0; j < NT; ++j) {
      v2f b;
      b[0] = Bcol[(long)ka * DIM + j * 16];
      b[1] = Bcol[(long)(ka + 1) * DIM + j * 16];
      // 8 args: (neg_a, A, neg_b, B, c_mod, C, reuse_a, reuse_b)
      c[j] = __builtin_amdgcn_wmma_f32_16x16x4_f32(false, a, false, b,
                                                   (short)0, c[j], false, false);
    }
  }

#pragma unroll
  for (int j = 0; j < NT; ++j) {
    float bv = (MODE == 0) ? bias[nbase + j * 16 + lm] : 0.0f;
#pragma unroll
    for (int i = 0; i < 8; ++i) {
      int row = m0 + i + hi * 8;
      if (MODE == 2 && row >= M) continue;
      C[(long)row * DIM + nbase + j * 16 + lm] = c[j][i] + bv;
    }
  }
}

// Per-column sum / sum-of-squares; thread t owns column t (coalesced rows).
__global__ void bnstats_kernel(const float* __restrict__ H, float* __restrict__ sums) {
  int col = threadIdx.x;  // 256 threads
  float s = 0.f, s2 = 0.f;
  for (int row = blockIdx.x; row < N_NODES; row += gridDim.x) {
    float v = H[(long)row * DIM + col];
    s += v; s2 += v * v;
  }
  unsafeAtomicAdd(&sums[col], s);
  unsafeAtomicAdd(&sums[DIM + col], s2);
}

__global__ void bnfinal_kernel(float* __restrict__ sums, const float* __restrict__ gamma,
                               const float* __restrict__ beta) {
  int c = threadIdx.x;
  float inv_n  = 1.0f / (float)N_NODES;
  float mean   = sums[c] * inv_n;
  float var    = sums[DIM + c] * inv_n - mean * mean;
  float scale  = gamma[c] * rsqrtf(var + BN_EPS);
  sums[2 * DIM + c] = scale;
  sums[3 * DIM + c] = beta[c] - mean * scale;
}

__global__ void normtanh_kernel(float* __restrict__ H, const float* __restrict__ sums) {
  long i = (long)blockIdx.x * blockDim.x + threadIdx.x;
  if (i >= (long)N_NODES * DIM) return;
  int c = (int)(i & (DIM - 1));
  H[i] = tanhf(H[i] * sums[2 * DIM + c] + sums[3 * DIM + c]);
}

extern "C" void kernel_launch(void* const* d_in, const int* in_sizes, int n_in,
                              void* d_out, int out_size, void* d_ws, size_t ws_size,
                              hipStream_t stream) {
  const float* x        = (const float*)d_in[0];
  const float* r        = (const float*)d_in[1];
  const float* w_in     = (const float*)d_in[2];
  const float* w_out    = (const float*)d_in[3];
  const float* w_loop   = (const float*)d_in[4];
  const float* w_rel    = (const float*)d_in[5];
  const float* loop_rel = (const float*)d_in[6];
  const float* bias     = (const float*)d_in[7];
  const float* gamma    = (const float*)d_in[8];
  const float* beta     = (const float*)d_in[9];
  const int* edge_index = (const int*)d_in[10];
  const int* edge_type  = (const int*)d_in[11];

  float* H     = (float*)d_out;                 // [N, D] node output
  float* r_out = H + (long)N_NODES * DIM;       // [R, D] relation output
  float* acc   = (float*)d_ws;                  // [N, D] scatter accumulator
  float* deg   = acc + (long)N_NODES * DIM;     // [N] degree -> deg^-0.5 in place
  float* sums  = deg + N_NODES;                 // [4*D] sum|sumsq|scale|shift

  const long accn = (long)N_NODES * DIM;
  const int half = N_EDGES / 2;
  // main GEMM: 6250 m-tiles * 4 n-groups = 25000 waves / 8 waves-per-block
  const int gemm_blocks = (N_NODES / 16) * (DIM / (16 * NT)) / 8;   // 3125
  const long sthreads = (long)half * 64;                            // 64 thr/edge
  const int scatter_blocks = (int)((sthreads + 255) / 256);

  // zero acc + deg + sums (contiguous)
  zero_f32<<<2048, 256, 0, stream>>>(acc, accn + N_NODES + 4 * DIM);
  // degrees over all E edges, then deg^-0.5
  degree_kernel<<<(N_EDGES + 255) / 256, 256, 0, stream>>>(edge_index, deg, N_EDGES);
  deginv_kernel<<<(N_NODES + 255) / 256, 256, 0, stream>>>(deg, N_NODES);

  // H = (x - loop_rel) @ w_loop + bias
  gemm_kernel<0><<<gemm_blocks, 256, 0, stream>>>(x, w_loop, loop_rel, bias, H, N_NODES);

  // pass 1: 'in' half
  scatter_kernel<<<scatter_blocks, 256, 0, stream>>>(x, r, edge_index, edge_type, deg, acc, 0, half);
  gemm_kernel<1><<<gemm_blocks, 256, 0, stream>>>(acc, w_in, nullptr, nullptr, H, N_NODES);

  // pass 2: 'out' half (reuse acc)
  zero_f32<<<2048, 256, 0, stream>>>(acc, accn);
  scatter_kernel<<<scatter_blocks, 256, 0, stream>>>(x, r, edge_index, edge_type, deg, acc, half, half);
  gemm_kernel<1><<<gemm_blocks, 256, 0, stream>>>(acc, w_out, nullptr, nullptr, H, N_NODES);

  // batchnorm over node dim + tanh, in place in d_out
  bnstats_kernel<<<512, 256, 0, stream>>>(H, sums);
  bnfinal_kernel<<<1, 256, 0, stream>>>(sums, gamma, beta);
  normtanh_kernel<<<(int)((accn + 255) / 256), 256, 0, stream>>>(H, sums);

  // r_out = r @ w_rel : ceil(474/16)=30 m-tiles * 4 n-groups = 120 waves / 8 = 15 blocks
  gemm_kernel<2><<<15, 256, 0, stream>>>(r, w_rel, nullptr, nullptr, r_out, N_REL);
}